// DNN_module_29420525977905
// MI455X (gfx1250) — compile-verified
//
#include <hip/hip_runtime.h>
#include <hip/hip_bf16.h>
#include <math.h>

typedef __attribute__((ext_vector_type(2))) float v2f;
typedef __attribute__((ext_vector_type(8))) float v8f;
typedef __attribute__((ext_vector_type(4))) int   v4i;

typedef __attribute__((address_space(1))) v4i* gv4i_ptr;  // global (AS1)
typedef __attribute__((address_space(3))) v4i* lv4i_ptr;  // LDS (AS3)

#define KNN 30
#define LCAP 32      // per-(wave,row,half) sorted candidate list capacity
#define NWAVE 8      // waves per workgroup
#define NMAX 10016   // capacity of the LDS-staged point array (n=10000)

#if __has_builtin(__builtin_amdgcn_global_load_async_to_lds_b128) && \
    __has_builtin(__builtin_amdgcn_s_wait_asynccnt)
#define USE_ASYNC_LDS 1
#endif

// --------------------------------------------------------------------------
// Kernel 1: one workgroup computes mean + covariance (ddof=1), inverts the
// 3x3, takes its Cholesky factor L (lower, L L^T = inv(cov)), writes
// ws[0..5] = L00,L10,L20,L11,L21,L22 ; ws[6..8] = mean.
// --------------------------------------------------------------------------
__global__ void stats_kernel(const float* __restrict__ c, const float* __restrict__ u,
                             const float* __restrict__ s, float* __restrict__ wsf, int n) {
  __shared__ float red[256];
  __shared__ float res[9];
  int tid = threadIdx.x;
  float a[9] = {0, 0, 0, 0, 0, 0, 0, 0, 0};
  for (int i = tid; i < n; i += 256) {
    float x = c[i], y = u[i], z = s[i];
    a[0] += x;     a[1] += y;     a[2] += z;
    a[3] += x * x; a[4] += x * y; a[5] += x * z;
    a[6] += y * y; a[7] += y * z; a[8] += z * z;
  }
  for (int q = 0; q < 9; ++q) {
    red[tid] = a[q];
    __syncthreads();
    for (int off = 128; off > 0; off >>= 1) {
      if (tid < off) red[tid] += red[tid + off];
      __syncthreads();
    }
    if (tid == 0) res[q] = red[0];
    __syncthreads();
  }
  if (tid == 0) {
    float fn = (float)n;
    float m0 = res[0] / fn, m1 = res[1] / fn, m2 = res[2] / fn;
    float inv = 1.0f / (fn - 1.0f);
    float caa = (res[3] - fn * m0 * m0) * inv;
    float cab = (res[4] - fn * m0 * m1) * inv;
    float cac = (res[5] - fn * m0 * m2) * inv;
    float cbb = (res[6] - fn * m1 * m1) * inv;
    float cbc = (res[7] - fn * m1 * m2) * inv;
    float ccc = (res[8] - fn * m2 * m2) * inv;
    float det = caa * (cbb * ccc - cbc * cbc)
              - cab * (cab * ccc - cbc * cac)
              + cac * (cab * cbc - cbb * cac);
    float id = 1.0f / det;
    float V00 = (cbb * ccc - cbc * cbc) * id;
    float V01 = (cac * cbc - cab * ccc) * id;
    float V02 = (cab * cbc - cac * cbb) * id;
    float V11 = (caa * ccc - cac * cac) * id;
    float V12 = (cab * cac - caa * cbc) * id;
    float V22 = (caa * cbb - cab * cab) * id;
    float L00 = sqrtf(V00);
    float L10 = V01 / L00;
    float L20 = V02 / L00;
    float L11 = sqrtf(V11 - L10 * L10);
    float L21 = (V12 - L20 * L10) / L11;
    float L22 = sqrtf(V22 - L20 * L20 - L21 * L21);
    wsf[0] = L00; wsf[1] = L10; wsf[2] = L20;
    wsf[3] = L11; wsf[4] = L21; wsf[5] = L22;
    wsf[6] = m0;  wsf[7] = m1;  wsf[8] = m2;
  }
}

// --------------------------------------------------------------------------
// Kernel 2: whiten. y = (p - mean) @ L (L lower triangular).
// Store float4 (y0, y1, y2, ||y||^2) at wsf+16 (16B aligned).
// --------------------------------------------------------------------------
__global__ void whiten_kernel(const float* __restrict__ c, const float* __restrict__ u,
                              const float* __restrict__ s, float* __restrict__ wsf, int n) {
  int i = blockIdx.x * blockDim.x + threadIdx.x;
  if (i >= n) return;
  float L00 = wsf[0], L10 = wsf[1], L20 = wsf[2];
  float L11 = wsf[3], L21 = wsf[4], L22 = wsf[5];
  float m0 = wsf[6], m1 = wsf[7], m2 = wsf[8];
  float x0 = c[i] - m0, x1 = u[i] - m1, x2 = s[i] - m2;
  float y0 = x0 * L00 + x1 * L10 + x2 * L20;
  float y1 = x1 * L11 + x2 * L21;
  float y2 = x2 * L22;
  float sq = y0 * y0 + y1 * y1 + y2 * y2;
  float4* y4 = (float4*)(wsf + 16);
  y4[i] = make_float4(y0, y1, y2, sq);
}

// --------------------------------------------------------------------------
// Kernel 3: 8 waves per workgroup, one workgroup per 16-row tile.
//  - Stage all n whitened points into LDS via async global->LDS (ASYNCcnt),
//    overlapped with candidate-list init; then the inner loop is fully
//    LDS-resident with zero barriers.
//  - Waves partition the 625 column tiles; each tile is one
//    V_WMMA_F32_16X16X4_F32. B fragments are software-pipelined one tile
//    ahead so the ds_load latency hides under the selection phase.
//  - Per-lane sorted top-32 lists with a register-resident rejection
//    threshold; 16-way merge per row at the end yields the top-30.
// --------------------------------------------------------------------------
__global__ void __launch_bounds__(256) knn_kernel(const float* __restrict__ wsf,
                                                  float* __restrict__ out, int n) {
  const float4* __restrict__ y4g = (const float4*)(wsf + 16);

  __shared__ float4 ylds[NMAX];                 // 160,256 B staged points
  __shared__ float  dtile[NWAVE][16][17];       // per-wave d2 tile (padded)
  __shared__ float  ldist[NWAVE][32][LCAP + 1]; // sorted candidate lists
  __shared__ int    lidx [NWAVE][32][LCAP + 1];

  const int tid  = threadIdx.x;
  const int wave = tid >> 5;
  const int lane = tid & 31;
  const int half = lane >> 4;     // 0: K=0,1 / rows 0-7 ; 1: K=2,pad / rows 8-15
  const int lr   = lane & 15;
  const int row_base = blockIdx.x * 16;
  const int ntiles = (n + 15) >> 4;
  const int nclamp = (n < NMAX) ? n : NMAX;

  // ---- stage y into LDS (async global->LDS when available) ----
#ifdef USE_ASYNC_LDS
  for (int i = tid; i < nclamp; i += 256) {
    __builtin_amdgcn_global_load_async_to_lds_b128(
        (gv4i_ptr)(y4g + i),            // global src (AS1, v4i*)
        (lv4i_ptr)(&ylds[i]),           // LDS dst (AS3, v4i*)
        /*offset=*/0, /*cpol=*/0);
  }
#else
  for (int i = tid; i < nclamp; i += 256) ylds[i] = y4g[i];
#endif

  // init candidate lists while the staging DMA is in flight (disjoint LDS)
  for (int k = 0; k < LCAP; ++k) {
    ldist[wave][lane][k] = 3.0e38f;
    lidx [wave][lane][k] = 0;
  }

#ifdef USE_ASYNC_LDS
  __builtin_amdgcn_s_wait_asynccnt(0);
#endif
  __syncthreads();   // staged y + all lists visible to all waves

  // A fragment (16x4 f32): lanes 0-15 hold K=0,1 ; lanes 16-31 hold K=2,3(=0)
  int ra = row_base + lr; if (ra > n - 1) ra = n - 1;
  float4 pa = ylds[ra];
  v2f afrag;
  afrag.x = half ? pa.z : pa.x;
  afrag.y = half ? 0.0f : pa.y;

  // squared norms of my 8 D-matrix rows (M = v + 8*half)
  float rsq[8];
#pragma unroll
  for (int v = 0; v < 8; ++v) {
    int rr = row_base + 8 * half + v; if (rr > n - 1) rr = n - 1;
    rsq[v] = ylds[rr].w;
  }

  float* myd = &ldist[wave][lane][0];
  int*   myi = &lidx [wave][lane][0];
  float worst = 3.0e38f;                // register-resident rejection threshold

  // prime the B-fragment pipeline with this wave's first tile
  int cc0 = (wave << 4) + lr; if (cc0 > n - 1) cc0 = n - 1;
  float4 pb = (wave < ntiles) ? ylds[cc0] : make_float4(0.f, 0.f, 0.f, 0.f);

  // ---- column tiles, strided across the 8 waves; no barriers inside ----
  for (int t = wave; t < ntiles; t += NWAVE) {
    const int cb = t << 4;
    float4 cur = pb;

    // software pipeline: issue next tile's LDS load now; its wait retires
    // under the selection phase below (uniform branch per wave)
    int tn = t + NWAVE;
    if (tn < ntiles) {
      int cc = (tn << 4) + lr; if (cc > n - 1) cc = n - 1;
      pb = ylds[cc];
    }

    v2f bfrag;                          // B mirrors A's layout (dot symmetric)
    bfrag.x = half ? cur.z : cur.x;
    bfrag.y = half ? 0.0f : cur.y;
    float csq = cur.w;                  // sq of column N = lane&15

    v8f acc = {};
    acc = __builtin_amdgcn_wmma_f32_16x16x4_f32(
        /*neg_a=*/false, afrag, /*neg_b=*/false, bfrag,
        /*c_mod=*/(short)0, acc, /*reuse_a=*/false, /*reuse_b=*/false);

    union { v8f v; float f[8]; } uacc; uacc.v = acc;
#pragma unroll
    for (int v = 0; v < 8; ++v) {
      // lane holds (M = v + 8*half, N = lane&15)
      dtile[wave][8 * half + v][lr] = rsq[v] + csq - 2.0f * uacc.f[v];
    }
    // same-wave LDS ops are in order: scatter above is visible to the
    // gather below without any barrier.

#pragma unroll 1
    for (int j = 0; j < 8; ++j) {
      int cn = 8 * half + j;
      float d = dtile[wave][lr][cn];
      int ci = cb + cn;
      if (ci < n && d < worst) {
        int pos = LCAP - 1;
        while (pos > 0 && myd[pos - 1] > d) {
          myd[pos] = myd[pos - 1];
          myi[pos] = myi[pos - 1];
          --pos;
        }
        myd[pos] = d;
        myi[pos] = ci;
        worst = myd[LCAP - 1];          // refresh threshold only on insert
      }
    }
  }

  __syncthreads();   // all waves' lists final

  // ---- 16-way merge per row (16 lists: 8 waves x 2 halves) ----
  if (tid < 16) {
    int r = tid;
    int row = row_base + r;
    if (row < n) {
      int p[16];
#pragma unroll
      for (int q = 0; q < 16; ++q) p[q] = 0;
      float* outd = out + (size_t)row * KNN;
      float* outi = out + (size_t)n * KNN + (size_t)row * KNN;
      for (int k = 0; k < KNN; ++k) {
        float best = 3.4e38f;
        int bq = 0, bp = 0;
#pragma unroll
        for (int q = 0; q < 16; ++q) {  // p[] indexed by constants: stays in regs
          int w = q >> 1, h = q & 1;
          float d = ldist[w][h * 16 + r][p[q]];
          if (d < best) { best = d; bq = q; bp = p[q]; }
        }
        int ix = lidx[bq >> 1][(bq & 1) * 16 + r][bp];  // single idx load
#pragma unroll
        for (int q = 0; q < 16; ++q) p[q] += (q == bq) ? 1 : 0;
        float d = fmaxf(best, 0.0f);
        outd[k] = sqrtf(fmaxf(d, 1e-12f));
        outi[k] = (float)ix;            // indices output (cast to out dtype)
      }
    }
  }
}

// --------------------------------------------------------------------------
extern "C" void kernel_launch(void* const* d_in, const int* in_sizes, int n_in,
                              void* d_out, int out_size, void* d_ws, size_t ws_size,
                              hipStream_t stream) {
  const float* c = (const float*)d_in[0];
  const float* u = (const float*)d_in[1];
  const float* s = (const float*)d_in[2];
  float* wsf = (float*)d_ws;   // [0..8]: L + mean ; +16 floats: float4 y[n]
  float* out = (float*)d_out;  // [n*30] dists, then [n*30] indices-as-float
  int n = in_sizes[0];

  stats_kernel<<<1, 256, 0, stream>>>(c, u, s, wsf, n);
  int wb = (n + 255) / 256;
  whiten_kernel<<<wb, 256, 0, stream>>>(c, u, s, wsf, n);
  int nt = (n + 15) / 16;
  knn_kernel<<<nt, 256, 0, stream>>>(wsf, out, n);
}